// kMinkMaxPooling1D_53970559041791
// MI455X (gfx1250) — compile-verified
//
#include <hip/hip_runtime.h>
#include <stdint.h>

// Problem constants (from reference: x = [32, 8192, 256] f32)
#define B 32
#define L 8192
#define C 256

#define CTILE 32            // channels per block
#define PHASES 8            // waves per block; wave w handles rows l == w (mod 8)
#define NBUF 8              // tiles in flight per wave (2 async B128 ops each)
#define TILE_ROWS 8         // rows (of this wave's phase) per tile
#define BUF_BYTES 1024      // TILE_ROWS * 32ch * 4B
#define BUF_FLOATS (BUF_BYTES / 4)
#define WAVE_LDS (NBUF * BUF_BYTES)        // 8192 B per wave -> 8KB in flight
#define SMEM_BYTES (PHASES * WAVE_LDS)     // 65536 B; merge area aliases offset 0
#define TILES ((L / PHASES) / TILE_ROWS)   // 128 tiles per wave

// ---- CDNA5 async global->LDS (VGLOBAL, tracked with ASYNCcnt) ----
// Non-temporal: 256MiB stream-once input > 192MB L2, don't pollute far caches.
__device__ __forceinline__ void async_tile_load_b128(unsigned lds_byte_addr,
                                                     const void* gaddr) {
  asm volatile("global_load_async_to_lds_b128 %0, %1, off th:TH_LOAD_NT"
               :
               : "v"(lds_byte_addr), "v"(gaddr)
               : "memory");
}

#define WAIT_ASYNCCNT(n) asm volatile("s_wait_asynccnt " #n ::: "memory")
#define WAIT_DSCNT0()    asm volatile("s_wait_dscnt 0" ::: "memory")

// ---- sorted-triple insertion: 3 VALU ops each via v_med3 ----
__device__ __forceinline__ void push_min(float& m0, float& m1, float& m2, float v) {
  float a0 = fminf(m0, v);
  float a1 = __builtin_amdgcn_fmed3f(m0, m1, v);
  float a2 = __builtin_amdgcn_fmed3f(m1, m2, v);
  m0 = a0; m1 = a1; m2 = a2;
}
__device__ __forceinline__ void push_max(float& M0, float& M1, float& M2, float v) {
  float a0 = fmaxf(M0, v);
  float a1 = __builtin_amdgcn_fmed3f(M0, M1, v);
  float a2 = __builtin_amdgcn_fmed3f(M1, M2, v);
  M0 = a0; M1 = a1; M2 = a2;
}

extern "C" __global__ void __launch_bounds__(256, 1)
kminkmax_kernel(const float* __restrict__ x, float* __restrict__ out) {
  extern __shared__ float smemF[];  // only LDS in kernel -> dynamic base = offset 0

  const int t    = threadIdx.x;
  const int lane = t & 31;          // wave32
  const int w    = t >> 5;          // phase / wave id: 0..7
  const int b     = blockIdx.y;
  const int cbase = blockIdx.x * CTILE;

  // ---- load-side lane mapping: lane -> (row-in-issue k, 16B chunk c16) ----
  const int k   = lane >> 3;        // 0..3
  const int c16 = lane & 7;         // 0..7  (16B = 4 channels)
  // byte address of this lane's first 16B chunk (tile 0, issue 0)
  const char* laneBase = (const char*)x +
      (((size_t)b * L + (size_t)(w + 8 * k)) * C + (size_t)cbase + (size_t)c16 * 4) * 4;
  // one tile advances 8 phase-rows = 64 L-rows = 64*1024 B
  const size_t tileStride  = (size_t)TILE_ROWS * 8 * C * 4;   // 65536 B
  const size_t issueStride = tileStride / 2;                  // 32768 B (rows +4 of phase)
  const unsigned ldsLaneBase = (unsigned)(w * WAVE_LDS) + (unsigned)lane * 16u;

  // ---- prologue: fill the per-wave pipeline (16 async ops = 8KB in flight) ----
#pragma unroll
  for (int i = 0; i < NBUF; ++i) {
    const char* g = laneBase + (size_t)i * tileStride;
    unsigned    l = ldsLaneBase + (unsigned)i * BUF_BYTES;
    async_tile_load_b128(l,        (const void*)g);
    async_tile_load_b128(l + 512u, (const void*)(g + issueStride));
  }

  const float INF = __builtin_inff();
  float m0 = INF, m1 = INF, m2 = INF;        // 3 smallest, ascending
  float M0 = -INF, M1 = -INF, M2 = -INF;     // 3 largest, descending

  const unsigned waveF = (unsigned)(w * WAVE_LDS) / 4;

  for (int i = 0; i < TILES; ++i) {
    // 2 ops/tile, in-order completion: tile i landed <=> asynccnt <= 2*NBUF-2
    if (i + NBUF < TILES) { WAIT_ASYNCCNT(14); }
    else                  { WAIT_ASYNCCNT(0); }   // drain tail
    const unsigned bufF = waveF + (unsigned)(i & (NBUF - 1)) * BUF_FLOATS;
#pragma unroll
    for (int r = 0; r < TILE_ROWS; ++r) {
      float v = smemF[bufF + (unsigned)r * 32 + (unsigned)lane];  // channel = lane
      push_min(m0, m1, m2, v);
      push_max(M0, M1, M2, v);
    }
    if (i + NBUF < TILES) {
      WAIT_DSCNT0();  // LDS reads of this slot retired before DMA overwrites it
      const char* g = laneBase + (size_t)(i + NBUF) * tileStride;
      unsigned    l = ldsLaneBase + (unsigned)(i & (NBUF - 1)) * BUF_BYTES;
      async_tile_load_b128(l,        (const void*)g);
      async_tile_load_b128(l + 512u, (const void*)(g + issueStride));
    }
  }

  // ---- merge 8 phase-partials per channel (merge area aliases buffer LDS) ----
  __syncthreads();   // all waves drained (asynccnt==0) and done reading buffers
  {
    unsigned mb = (unsigned)(w * CTILE + lane) * 6;
    smemF[mb + 0] = m0; smemF[mb + 1] = m1; smemF[mb + 2] = m2;
    smemF[mb + 3] = M0; smemF[mb + 4] = M1; smemF[mb + 5] = M2;
  }
  __syncthreads();

  if (t < CTILE) {
    float r0 = INF, r1 = INF, r2 = INF;
    float s0 = -INF, s1 = -INF, s2 = -INF;
#pragma unroll
    for (int p = 0; p < PHASES; ++p) {
      unsigned q = (unsigned)(p * CTILE + t) * 6;
      push_min(r0, r1, r2, smemF[q + 0]);
      push_min(r0, r1, r2, smemF[q + 1]);
      push_min(r0, r1, r2, smemF[q + 2]);
      push_max(s0, s1, s2, smemF[q + 3]);
      push_max(s0, s1, s2, smemF[q + 4]);
      push_max(s0, s1, s2, smemF[q + 5]);
    }
    float* o = out + ((size_t)b * C + (size_t)(cbase + t)) * 6;
    o[0] = r0; o[1] = r1; o[2] = r2;   // kmin ascending
    o[3] = s0; o[4] = s1; o[5] = s2;   // kmax descending
  }
}

extern "C" void kernel_launch(void* const* d_in, const int* in_sizes, int n_in,
                              void* d_out, int out_size, void* d_ws, size_t ws_size,
                              hipStream_t stream) {
  (void)in_sizes; (void)n_in; (void)d_ws; (void)ws_size; (void)out_size;
  const float* x = (const float*)d_in[0];
  float* out = (float*)d_out;
  dim3 grid(C / CTILE, B, 1);   // (8, 32) = 256 blocks
  dim3 block(256, 1, 1);        // 8 waves (wave32)
  kminkmax_kernel<<<grid, block, SMEM_BYTES, stream>>>(x, out);
}